// NaMixedOp_901943132757
// MI455X (gfx1250) — compile-verified
//
#include <hip/hip_runtime.h>
#include <hip/hip_bf16.h>
#include <math.h>

#define NEG_SLOPE 0.2f

typedef __attribute__((ext_vector_type(2))) float v2f;
typedef __attribute__((ext_vector_type(8))) float v8f;

__device__ __forceinline__ float lrelu(float z) {
  return z > 0.0f ? z : NEG_SLOPE * z;
}
__device__ __forceinline__ float eluf(float z) {
  return z > 0.0f ? z : (expf(z) - 1.0f);
}

// ---------------- init: zero accumulators, deg=1 (self loop), ncnt=0 --------
__global__ void k_init(float* zero_base, size_t zcount, float* deg, float* ncnt, int N) {
  size_t i = (size_t)blockIdx.x * blockDim.x + threadIdx.x;
  if (i < zcount) zero_base[i] = 0.0f;
  if (i < (size_t)N) { deg[i] = 1.0f; ncnt[i] = 0.0f; }
}

// ---------------- b_src = W_gat @ a_src ; b_dst = W_gat @ a_dst -------------
__global__ void k_prevec(const float* __restrict__ W_gat,
                         const float* __restrict__ a_src,
                         const float* __restrict__ a_dst,
                         float* __restrict__ b_src, float* __restrict__ b_dst) {
  int k = threadIdx.x;  // 0..63
  float s = 0.0f, d = 0.0f;
  for (int j = 0; j < 64; ++j) {
    float w = W_gat[k * 64 + j];
    s += w * a_src[j];
    d += w * a_dst[j];
  }
  b_src[k] = s;
  b_dst[k] = d;
}

// ---------------- transpose the 5 weight matrices: Wt[n][k] = W[k][n] -------
// Makes WMMA B-fragments {W[kb][col], W[kb+1][col]} contiguous -> b64 loads.
__global__ void k_transpose(const float* __restrict__ W0, const float* __restrict__ W1,
                            const float* __restrict__ W2, const float* __restrict__ W3,
                            const float* __restrict__ W4, float* __restrict__ Wt) {
  int idx = blockIdx.x * blockDim.x + threadIdx.x;  // 0 .. 5*4096-1
  if (idx >= 5 * 4096) return;
  int m = idx >> 12;
  int r = idx & 4095;
  int k = r >> 6, n = r & 63;
  const float* W = (m == 0) ? W0 : (m == 1) ? W1 : (m == 2) ? W2 : (m == 3) ? W3 : W4;
  Wt[(size_t)m * 4096 + n * 64 + k] = W[k * 64 + n];
}

// ---------------- per-edge degree / neighbor count --------------------------
__global__ void k_deg(const int* __restrict__ dst, int E,
                      float* __restrict__ deg, float* __restrict__ ncnt) {
  int e = blockIdx.x * blockDim.x + threadIdx.x;
  if (e >= E) return;
  int d = dst[e];
  atomicAdd(&deg[d], 1.0f);
  atomicAdd(&ncnt[d], 1.0f);
}

// ---- per-node: dinv, alphas, esum init with self-loop softmax numerator ----
__global__ void k_node1(const float* __restrict__ x,
                        const float* __restrict__ b_src, const float* __restrict__ b_dst,
                        float* __restrict__ deg_dinv,
                        float* __restrict__ alpha_s, float* __restrict__ alpha_d,
                        float* __restrict__ esum, int N) {
  int i = blockIdx.x * blockDim.x + threadIdx.x;
  if (i >= N) return;
  deg_dinv[i] = rsqrtf(deg_dinv[i]);  // deg >= 1 (self-loop)
  const float* xr = x + (size_t)i * 64;
  float s = 0.0f, d = 0.0f;
  for (int k = 0; k < 64; ++k) {
    float xv = xr[k];
    s += xv * b_src[k];
    d += xv * b_dst[k];
  }
  alpha_s[i] = s;
  alpha_d[i] = d;
  // unshifted softmax numerator of the self-loop edge (shift-invariant softmax)
  esum[i] = expf(lrelu(s + d));
}

// ------- single fused edge pass: gcn_pre, nsum, gat_pre scatters + esum -----
// One wave32 per edge; lane k handles components k and k+32 of the 64-dim row.
__global__ void k_edge(const int* __restrict__ src, const int* __restrict__ dst, int E,
                       const float* __restrict__ x, const float* __restrict__ dinv,
                       const float* __restrict__ alpha_s, const float* __restrict__ alpha_d,
                       float* __restrict__ gcn_pre, float* __restrict__ nsum,
                       float* __restrict__ gat_pre, float* __restrict__ esum) {
  int wid  = (blockIdx.x * blockDim.x + threadIdx.x) >> 5;
  int lane = threadIdx.x & 31;
  if (wid >= E) return;
  int s = src[wid], d = dst[wid];
  float dv = dinv[s];
  float ee = expf(lrelu(alpha_s[s] + alpha_d[d]));  // unshifted numerator
  const float* xs = x + (size_t)s * 64;
  float* gp = gcn_pre + (size_t)d * 64;
  float* np = nsum + (size_t)d * 64;
  float* gt = gat_pre + (size_t)d * 64;
  float x0 = xs[lane], x1 = xs[lane + 32];
  atomicAdd(&gp[lane],      dv * x0);
  atomicAdd(&gp[lane + 32], dv * x1);
  atomicAdd(&np[lane],      x0);
  atomicAdd(&np[lane + 32], x1);
  atomicAdd(&gt[lane],      ee * x0);
  atomicAdd(&gt[lane + 32], ee * x1);
  if (lane == 0) atomicAdd(&esum[d], ee);
}

// ---------------- fused WMMA epilogue: 5 GEMMs + ELU + weighted combine -----
// grid: N/16 blocks (N divisible by 16), block: 128 threads = 4 waves.
// Wave w handles output columns [16w, 16w+16). B read from transposed weights.
__global__ __launch_bounds__(128) void k_final(
    const float* __restrict__ x, const float* __restrict__ weights,
    const float* __restrict__ Wt,  // 5 transposed 64x64 matrices, 4096 floats apart
    const float* __restrict__ gcn_pre, const float* __restrict__ nsum,
    const float* __restrict__ gat_pre,
    const float* __restrict__ dinv, const float* __restrict__ ncnt,
    const float* __restrict__ alpha_s, const float* __restrict__ alpha_d,
    const float* __restrict__ esum,
    float* __restrict__ out) {
  int lane = threadIdx.x & 31;
  int wave = threadIdx.x >> 5;                  // column tile 0..3
  int r16  = lane & 15;
  int hi   = (lane >= 16) ? 1 : 0;
  int row  = blockIdx.x * 16 + r16;             // A-matrix row this lane feeds
  int col  = wave * 16 + r16;                   // B/C/D column this lane owns

  // per-row scalars
  float dv      = dinv[row];
  float invc    = 1.0f / fmaxf(ncnt[row], 1.0f);
  float ee_self = expf(lrelu(alpha_s[row] + alpha_d[row]));
  float inv_es  = 1.0f / esum[row];

  const float* xr  = x       + (size_t)row * 64;
  const float* gpr = gcn_pre + (size_t)row * 64;
  const float* nsr = nsum    + (size_t)row * 64;
  const float* gtr = gat_pre + (size_t)row * 64;

  // transposed B columns for this lane (contiguous along K -> b64 loads)
  const float* bt_gcn   = Wt + 0 * 4096 + (size_t)col * 64;
  const float* bt_self  = Wt + 1 * 4096 + (size_t)col * 64;
  const float* bt_neigh = Wt + 2 * 4096 + (size_t)col * 64;
  const float* bt_gin   = Wt + 3 * 4096 + (size_t)col * 64;
  const float* bt_gat   = Wt + 4 * 4096 + (size_t)col * 64;

  v8f acc_gcn = {}, acc_self = {}, acc_neigh = {}, acc_gin = {}, acc_gat = {};

#pragma unroll 4
  for (int kk = 0; kk < 16; ++kk) {
    int kb = 4 * kk + (hi ? 2 : 0);  // lanes 0-15: K=k0,k0+1 ; lanes 16-31: K=k0+2,k0+3

    v2f xv = *(const v2f*)(xr + kb);
    v2f ns = *(const v2f*)(nsr + kb);
    v2f gp = *(const v2f*)(gpr + kb);
    v2f gt = *(const v2f*)(gtr + kb);

    v2f a_gcn   = { dv * (gp.x + dv * xv.x),            dv * (gp.y + dv * xv.y) };
    v2f a_self  = xv;
    v2f a_neigh = { ns.x * invc,                        ns.y * invc };
    v2f a_gin   = { xv.x + ns.x,                        xv.y + ns.y };
    v2f a_gat   = { (gt.x + ee_self * xv.x) * inv_es,   (gt.y + ee_self * xv.y) * inv_es };

    v2f b_gcn   = *(const v2f*)(bt_gcn + kb);
    v2f b_self  = *(const v2f*)(bt_self + kb);
    v2f b_neigh = *(const v2f*)(bt_neigh + kb);
    v2f b_gin   = *(const v2f*)(bt_gin + kb);
    v2f b_gat   = *(const v2f*)(bt_gat + kb);

    acc_gcn   = __builtin_amdgcn_wmma_f32_16x16x4_f32(false, a_gcn,   false, b_gcn,   (short)0, acc_gcn,   false, false);
    acc_self  = __builtin_amdgcn_wmma_f32_16x16x4_f32(false, a_self,  false, b_self,  (short)0, acc_self,  false, false);
    acc_neigh = __builtin_amdgcn_wmma_f32_16x16x4_f32(false, a_neigh, false, b_neigh, (short)0, acc_neigh, false, false);
    acc_gin   = __builtin_amdgcn_wmma_f32_16x16x4_f32(false, a_gin,   false, b_gin,   (short)0, acc_gin,   false, false);
    acc_gat   = __builtin_amdgcn_wmma_f32_16x16x4_f32(false, a_gat,   false, b_gat,   (short)0, acc_gat,   false, false);
  }

  float w0 = weights[0], w1 = weights[1], w2 = weights[2], w3 = weights[3];
  int mbase = blockIdx.x * 16 + (hi ? 8 : 0);   // C/D layout: VGPR v -> row v (+8 for hi lanes)
#pragma unroll
  for (int v = 0; v < 8; ++v) {
    float sage = acc_self[v] + acc_neigh[v];
    float val = w0 * eluf(acc_gcn[v]) + w1 * eluf(sage) +
                w2 * eluf(acc_gin[v]) + w3 * eluf(acc_gat[v]);
    out[(size_t)(mbase + v) * 64 + col] = val;
  }
}

extern "C" void kernel_launch(void* const* d_in, const int* in_sizes, int n_in,
                              void* d_out, int out_size, void* d_ws, size_t ws_size,
                              hipStream_t stream) {
  const float* x       = (const float*)d_in[0];
  const float* weights = (const float*)d_in[1];
  const int*   edge    = (const int*)d_in[2];
  const float* W_gcn   = (const float*)d_in[3];
  const float* W_self  = (const float*)d_in[4];
  const float* W_neigh = (const float*)d_in[5];
  const float* W_gin   = (const float*)d_in[6];
  const float* W_gat   = (const float*)d_in[7];
  const float* a_src   = (const float*)d_in[8];
  const float* a_dst   = (const float*)d_in[9];

  int N = in_sizes[0] / 64;
  int E = in_sizes[2] / 2;
  const int* srcI = edge;
  const int* dstI = edge + E;

  // workspace layout (floats)
  float* ws = (float*)d_ws;
  float* gcn_pre = ws; ws += (size_t)N * 64;
  float* nsum    = ws; ws += (size_t)N * 64;
  float* gat_pre = ws; ws += (size_t)N * 64;
  float* deg     = ws; ws += N;   // becomes dinv after k_node1
  float* ncnt    = ws; ws += N;
  float* alpha_s = ws; ws += N;
  float* alpha_d = ws; ws += N;
  float* esum    = ws; ws += N;
  float* b_src   = ws; ws += 64;
  float* b_dst   = ws; ws += 64;
  float* Wt      = ws; ws += 5 * 4096;

  size_t zcount = (size_t)3 * N * 64;
  int blkInit = (int)((zcount + 255) / 256);
  k_init<<<blkInit, 256, 0, stream>>>(gcn_pre, zcount, deg, ncnt, N);
  k_prevec<<<1, 64, 0, stream>>>(W_gat, a_src, a_dst, b_src, b_dst);
  k_transpose<<<(5 * 4096 + 255) / 256, 256, 0, stream>>>(W_gcn, W_self, W_neigh,
                                                          W_gin, W_gat, Wt);
  k_deg<<<(E + 255) / 256, 256, 0, stream>>>(dstI, E, deg, ncnt);
  k_node1<<<(N + 255) / 256, 256, 0, stream>>>(x, b_src, b_dst, deg, alpha_s, alpha_d,
                                               esum, N);
  k_edge<<<(E + 7) / 8, 256, 0, stream>>>(srcI, dstI, E, x, deg, alpha_s, alpha_d,
                                          gcn_pre, nsum, gat_pre, esum);
  k_final<<<N / 16, 128, 0, stream>>>(x, weights, Wt, gcn_pre, nsum, gat_pre,
                                      deg, ncnt, alpha_s, alpha_d, esum, (float*)d_out);
}